// MixtralRouter_30262339567729
// MI455X (gfx1250) — compile-verified
//
#include <hip/hip_runtime.h>
#include <math.h>

// MoE router for MI455X (gfx1250, wave32).
// logits = hidden(32768,4096) @ gate_w^T(4096,8); bias; /clip(temp); top-2; softmax.
// Memory-bound: 512MB of hidden_states @ 23.3TB/s ~ 23us floor.
// Uses V_WMMA_F32_16X16X4_F32 (native fp32 matrix op) with gate_w resident in LDS.

typedef float v2f __attribute__((ext_vector_type(2)));
typedef float v8f __attribute__((ext_vector_type(8)));

#define D_DIM 4096
#define E_DIM 8
#define TOK_PER_WAVE 16
#define WAVES_PER_BLOCK 8
#define TOK_PER_BLOCK (TOK_PER_WAVE * WAVES_PER_BLOCK)   // 128
#define GW_FLOATS (E_DIM * D_DIM)                        // 32768 floats = 128 KB
#define STAGE_FLOATS (TOK_PER_BLOCK * E_DIM)             // 1024 floats = 4 KB
#define LDS_BYTES ((GW_FLOATS + STAGE_FLOATS) * 4)       // 132 KB dynamic LDS

__global__ __launch_bounds__(256, 2)
void router_wmma_f32_kernel(const float* __restrict__ hidden,
                            const float* __restrict__ pressure_bias,
                            const float* __restrict__ temperature,
                            const float* __restrict__ gate_w,
                            float* __restrict__ out_w,   // (tokens,2) routing weights
                            int*   __restrict__ out_e)   // (tokens,2) expert ids
{
    extern __shared__ float lds[];
    float* gw    = lds;               // 8 x 4096 fp32 gate weights
    float* stage = lds + GW_FLOATS;   // per-block logit staging

    const int tid = threadIdx.x;

    // ---- cooperative preload of gate_w into LDS (b128 copies) ----
    {
        const float4* src = (const float4*)gate_w;
        float4*       dst = (float4*)gw;
        #pragma unroll 4
        for (int i = tid; i < GW_FLOATS / 4; i += 256) dst[i] = src[i];
    }
    __syncthreads();

    const int wave = tid >> 5;
    const int lane = tid & 31;
    const int m    = lane & 15;   // row (A) / col (B) index within tile
    const int h    = lane >> 4;   // half-wave selects K pair: K = 2h + {0,1}
    const int tile_base = (blockIdx.x * WAVES_PER_BLOCK + wave) * TOK_PER_WAVE;
    const int tok  = tile_base + m;

    // A: lane L holds row M=L%16, K pair 2*(L/16)+{0,1}  (32-bit A 16x4 layout)
    const float* aptr = hidden + (size_t)tok * D_DIM + 2 * h;
    // B: lane L holds col N=L%16, same K pair. Cols 8..15 of C are discarded,
    // so clamp the expert index to stay in-bounds; values are don't-care.
    const float* bptr = gw + (m & (E_DIM - 1)) * D_DIM + 2 * h;

    v8f c = {0.f, 0.f, 0.f, 0.f, 0.f, 0.f, 0.f, 0.f};

    // K loop: 32 elements (= 8 WMMAs of K=4) per iteration.
    // Per iteration each token row consumes exactly one 128B cache line.
    for (int k0 = 0; k0 < D_DIM; k0 += 32) {
        v2f a[8], b[8];
        #pragma unroll
        for (int j = 0; j < 8; ++j) {
            a[j] = *(const v2f*)(aptr + k0 + 4 * j);   // global_load_b64
            b[j] = *(const v2f*)(bptr + k0 + 4 * j);   // ds_load_b64
        }
        __builtin_prefetch(aptr + k0 + 512, 0, 1);     // global_prefetch_b8, ~16 iters ahead
        #pragma unroll
        for (int j = 0; j < 8; ++j) {
            // D = A(16x4) * B(4x16) + C(16x16), fp32
            c = __builtin_amdgcn_wmma_f32_16x16x4_f32(
                    /*neg_a=*/false, a[j], /*neg_b=*/false, b[j],
                    /*c_mod=*/(short)0, c, /*reuse_a=*/false, /*reuse_b=*/false);
        }
    }

    // ---- stage C to LDS: VGPR r of half h is token (r+8h), expert m ----
    if (m < E_DIM) {
        #pragma unroll
        for (int r = 0; r < 8; ++r) {
            int tl = r + 8 * h;   // token within wave tile, 0..15
            stage[(wave * TOK_PER_WAVE + tl) * E_DIM + m] = c[r];
        }
    }
    __syncthreads();

    // ---- epilogue: one token per lane (lanes 0..15 of each wave) ----
    if (lane < TOK_PER_WAVE) {
        const float* lg = &stage[(wave * TOK_PER_WAVE + lane) * E_DIM];
        float v0 = -INFINITY, v1 = -INFINITY;
        int   i0 = 0,         i1 = 0;
        #pragma unroll
        for (int e = 0; e < E_DIM; ++e) {
            float t = temperature[e];
            t = fminf(fmaxf(t, 0.1f), 10.0f);
            float v = (lg[e] + pressure_bias[e]) / t;
            if (v > v0)      { v1 = v0; i1 = i0; v0 = v; i0 = e; }  // strict >: lower
            else if (v > v1) { v1 = v;  i1 = e; }                   // index wins ties
        }
        // softmax over the two selected logits (stable: shift by v0)
        float e1 = __expf(v1 - v0);
        float w0 = 1.0f / (1.0f + e1);
        float w1 = e1 * w0;

        const int token = tile_base + lane;
        out_w[token * 2 + 0] = w0;
        out_w[token * 2 + 1] = w1;
        out_e[token * 2 + 0] = i0;
        out_e[token * 2 + 1] = i1;
    }
}

extern "C" void kernel_launch(void* const* d_in, const int* in_sizes, int n_in,
                              void* d_out, int out_size, void* d_ws, size_t ws_size,
                              hipStream_t stream) {
    (void)n_in; (void)out_size; (void)d_ws; (void)ws_size;
    const float* hidden = (const float*)d_in[0];   // (B,S,D) fp32
    const float* bias   = (const float*)d_in[1];   // (E,)    fp32
    const float* temp   = (const float*)d_in[2];   // (E,)    fp32
    const float* gate_w = (const float*)d_in[3];   // (E,D)   fp32

    const int n_tokens = in_sizes[0] / D_DIM;      // 32768

    // Output tuple concatenated flat: weights (n_tokens*2 f32) then ids (n_tokens*2 i32).
    float* out_w = (float*)d_out;
    int*   out_e = (int*)(out_w + (size_t)n_tokens * 2);

    const int blocks = n_tokens / TOK_PER_BLOCK;   // 256
    router_wmma_f32_kernel<<<dim3(blocks), dim3(256), LDS_BYTES, stream>>>(
        hidden, bias, temp, gate_w, out_w, out_e);
}